// GCN_model_87737591923138
// MI455X (gfx1250) — compile-verified
//
#include <hip/hip_runtime.h>

// GCN (2x GCNConv + mean-pool + log_softmax) for MI455X / gfx1250.
// Dense linear layers use V_WMMA_F32_16X16X4_F32 (exact f32 accumulate) with a
// branch-free inner loop: A via global_load_b64, B via a single ds_load_b64
// from a zero-padded, lane-packed LDS image of the weights.
// Edge aggregation is scatter-add with global_atomic_add_f32 (L2 resident).
// Pooling pre-reduces in LDS (ds_add_f32) before global atomics.

typedef __attribute__((ext_vector_type(2))) float v2f;
typedef __attribute__((ext_vector_type(8))) float v8f;

#define NN 200000
#define EE 6400000
#define FF 128
#define HH 6
#define CC 10
#define GG 64

// ---------------------------------------------------------------- init / degree
__global__ __launch_bounds__(256) void init_kernel(float* __restrict__ deg,
                                                   float* __restrict__ gsum,
                                                   float* __restrict__ gcnt) {
    int i = blockIdx.x * 256 + threadIdx.x;
    if (i < NN) deg[i] = 1.0f;            // +1 for self loop
    if (i < GG * CC) gsum[i] = 0.0f;
    if (i < GG) gcnt[i] = 0.0f;
}

__global__ __launch_bounds__(256) void deg_kernel(const int* __restrict__ col,
                                                  float* __restrict__ deg) {
    int e = blockIdx.x * 256 + threadIdx.x;
    if (e < EE) atomicAdd(&deg[col[e]], 1.0f);
}

__global__ __launch_bounds__(256) void dis_kernel(float* __restrict__ deg) {
    int i = blockIdx.x * 256 + threadIdx.x;
    if (i < NN) deg[i] = rsqrtf(deg[i]);  // in-place: deg -> deg^{-1/2}
}

// ---------------------------------------------------------------- gemm1 (WMMA)
// h1 = x @ W1 ;  agg1 = dis^2 * h1 + b1   (self-loop term pre-seeded)
// One wave per 16-row tile, K=128 in 32 steps of V_WMMA_F32_16X16X4_F32.
// B fragments packed in LDS as sB[kstep][col*4 + kq], zero-padded cols 6..15,
// so each lane's (K, K+1) pair is one contiguous 8-byte ds_load.
__global__ __launch_bounds__(128) void gemm1_kernel(const float* __restrict__ x,
                                                    const float* __restrict__ W1,
                                                    const float* __restrict__ b1,
                                                    const float* __restrict__ dis,
                                                    float* __restrict__ h1,
                                                    float* __restrict__ agg1) {
    __shared__ float sB[32 * 64];                 // [k/4][col*4 + kq], 8 KB
    for (int t = threadIdx.x; t < 32 * 64; t += 128) {
        int k4 = t >> 6, rem = t & 63, n = rem >> 2, kq = rem & 3;
        sB[t] = (n < HH) ? W1[(k4 * 4 + kq) * HH + n] : 0.0f;
    }
    __syncthreads();

    const int lane  = threadIdx.x & 31;
    const int tile  = blockIdx.x * 4 + (threadIdx.x >> 5);   // 12500 tiles, exact
    const int m     = lane & 15;                              // A row within tile
    const int khalf = (lane >> 4) << 1;                       // 0 or 2 (K sub-offset)
    const int colv  = lane & 15;                              // B/D column

    const float* xrow = x + (size_t)(tile * 16 + m) * FF + khalf;
    const float* bptr = &sB[colv * 4 + khalf];                // lane-fixed base

    v8f acc = {};
#pragma unroll
    for (int k4 = 0; k4 < 32; ++k4) {
        v2f a = *(const v2f*)(xrow + k4 * 4);                 // A: K, K+1
        v2f b = *(const v2f*)(bptr + k4 * 64);                // B: K, K+1 (one b64)
        acc = __builtin_amdgcn_wmma_f32_16x16x4_f32(
            false, a, false, b, (short)0, acc, false, false);
    }

    if (colv < HH) {
        const int base = tile * 16 + ((lane >> 4) << 3);      // M = r + 8*(lane/16)
        const float bb = b1[colv];
#pragma unroll
        for (int r = 0; r < 8; ++r) {
            float v = acc[r];
            float d = dis[base + r];
            h1[(size_t)(base + r) * HH + colv]   = v;
            agg1[(size_t)(base + r) * HH + colv] = d * d * v + bb;
        }
    }
}

// ---------------------------------------------------------------- edge scatter 1
__global__ __launch_bounds__(256) void edge1_kernel(const int* __restrict__ row,
                                                    const int* __restrict__ col,
                                                    const float* __restrict__ dis,
                                                    const float* __restrict__ h1,
                                                    float* __restrict__ agg1) {
    int e = blockIdx.x * 256 + threadIdx.x;
    if (e >= EE) return;
    int r = row[e], c = col[e];
    float nrm = dis[r] * dis[c];
    const float* hr = h1 + (size_t)r * HH;
    float* ac = agg1 + (size_t)c * HH;
#pragma unroll
    for (int j = 0; j < HH; ++j) atomicAdd(ac + j, nrm * hr[j]);
}

// ---------------------------------------------------------------- gemm2 (WMMA)
// h2 = relu(agg1) @ W2 ; agg2 = dis^2 * h2 + b2. K=6 padded to 8 (2 WMMA steps).
// A padding via clamped loads + selects (no EXEC divergence at the WMMA).
__global__ __launch_bounds__(128) void gemm2_kernel(const float* __restrict__ agg1,
                                                    const float* __restrict__ W2,
                                                    const float* __restrict__ b2,
                                                    const float* __restrict__ dis,
                                                    float* __restrict__ h2,
                                                    float* __restrict__ agg2) {
    __shared__ float sB2[2 * 64];                 // [kstep][col*4 + kq], padded
    if (threadIdx.x < 2 * 64) {
        int t = threadIdx.x;
        int ks = t >> 6, rem = t & 63, n = rem >> 2, kq = rem & 3;
        int kk = ks * 4 + kq;
        sB2[t] = (kk < HH && n < CC) ? W2[kk * CC + n] : 0.0f;
    }
    __syncthreads();

    const int lane  = threadIdx.x & 31;
    const int tile  = blockIdx.x * 4 + (threadIdx.x >> 5);
    const int m     = lane & 15;
    const int khalf = (lane >> 4) << 1;
    const int colv  = lane & 15;
    const int arow  = tile * 16 + m;

    v8f acc = {};
#pragma unroll
    for (int ks = 0; ks < 2; ++ks) {
        int kk = ks * 4 + khalf;                              // 0,2,4,6 (even)
        int kc = (kk < HH) ? kk : 0;                          // clamp for safe load
        float ax = fmaxf(agg1[(size_t)arow * HH + kc], 0.0f);      // ReLU
        float ay = fmaxf(agg1[(size_t)arow * HH + kc + 1], 0.0f);  // kc+1 <= 5
        v2f a;
        a.x = (kk < HH) ? ax : 0.0f;                          // zero-pad K=6,7
        a.y = (kk < HH) ? ay : 0.0f;
        v2f b = *(const v2f*)(&sB2[ks * 64 + colv * 4 + khalf]);
        acc = __builtin_amdgcn_wmma_f32_16x16x4_f32(
            false, a, false, b, (short)0, acc, false, false);
    }

    if (colv < CC) {
        const int base = tile * 16 + ((lane >> 4) << 3);
        const float bb = b2[colv];
#pragma unroll
        for (int r = 0; r < 8; ++r) {
            float v = acc[r];
            float d = dis[base + r];
            h2[(size_t)(base + r) * CC + colv]   = v;
            agg2[(size_t)(base + r) * CC + colv] = d * d * v + bb;
        }
    }
}

// ---------------------------------------------------------------- edge scatter 2
__global__ __launch_bounds__(256) void edge2_kernel(const int* __restrict__ row,
                                                    const int* __restrict__ col,
                                                    const float* __restrict__ dis,
                                                    const float* __restrict__ h2,
                                                    float* __restrict__ agg2) {
    int e = blockIdx.x * 256 + threadIdx.x;
    if (e >= EE) return;
    int r = row[e], c = col[e];
    float nrm = dis[r] * dis[c];
    const float* hr = h2 + (size_t)r * CC;
    float* ac = agg2 + (size_t)c * CC;
#pragma unroll
    for (int j = 0; j < CC; ++j) atomicAdd(ac + j, nrm * hr[j]);
}

// ---------------------------------------------------------------- mean pool (LDS)
__global__ __launch_bounds__(256) void pool_kernel(const float* __restrict__ agg2,
                                                   const int* __restrict__ batch,
                                                   float* __restrict__ gsum,
                                                   float* __restrict__ gcnt) {
    __shared__ float ssum[GG * CC];
    __shared__ float scnt[GG];
    for (int t = threadIdx.x; t < GG * CC; t += 256) ssum[t] = 0.0f;
    if (threadIdx.x < GG) scnt[threadIdx.x] = 0.0f;
    __syncthreads();

    int i = blockIdx.x * 256 + threadIdx.x;
    if (i < NN) {
        int g = batch[i];
#pragma unroll
        for (int c = 0; c < CC; ++c)
            atomicAdd(&ssum[g * CC + c], agg2[(size_t)i * CC + c]);
        atomicAdd(&scnt[g], 1.0f);
    }
    __syncthreads();

    for (int t = threadIdx.x; t < GG * CC; t += 256) atomicAdd(&gsum[t], ssum[t]);
    if (threadIdx.x < GG) atomicAdd(&gcnt[threadIdx.x], scnt[threadIdx.x]);
}

// ---------------------------------------------------------------- log-softmax
__global__ __launch_bounds__(64) void lsm_kernel(const float* __restrict__ gsum,
                                                 const float* __restrict__ gcnt,
                                                 float* __restrict__ out) {
    int g = threadIdx.x;
    if (g >= GG) return;
    float inv = 1.0f / fmaxf(gcnt[g], 1.0f);
    float p[CC];
    float m = -3.0e38f;
#pragma unroll
    for (int c = 0; c < CC; ++c) { p[c] = gsum[g * CC + c] * inv; m = fmaxf(m, p[c]); }
    float s = 0.0f;
#pragma unroll
    for (int c = 0; c < CC; ++c) s += expf(p[c] - m);
    float l = logf(s);
#pragma unroll
    for (int c = 0; c < CC; ++c) out[g * CC + c] = p[c] - m - l;
}

// ---------------------------------------------------------------- launch
extern "C" void kernel_launch(void* const* d_in, const int* in_sizes, int n_in,
                              void* d_out, int out_size, void* d_ws, size_t ws_size,
                              hipStream_t stream) {
    const float* x     = (const float*)d_in[0];     // [N,128]
    const int*   eidx  = (const int*)d_in[1];       // [2,E]
    const int*   batch = (const int*)d_in[2];       // [N]
    const float* W1    = (const float*)d_in[3];     // [128,6]
    const float* b1    = (const float*)d_in[4];     // [6]
    const float* W2    = (const float*)d_in[5];     // [6,10]
    const float* b2    = (const float*)d_in[6];     // [10]
    float*       out   = (float*)d_out;             // [64,10]

    const int* erow = eidx;
    const int* ecol = eidx + EE;

    // Workspace layout (floats): 33N + 704 total (~26.4 MB)
    float* ws   = (float*)d_ws;
    float* dis  = ws;                        // [N]   degree -> deg^{-1/2} in place
    float* h1   = ws + (size_t)NN;           // [N,6]
    float* agg1 = h1 + (size_t)NN * HH;      // [N,6]
    float* h2   = agg1 + (size_t)NN * HH;    // [N,10]
    float* agg2 = h2 + (size_t)NN * CC;      // [N,10]
    float* gsum = agg2 + (size_t)NN * CC;    // [64,10]
    float* gcnt = gsum + GG * CC;            // [64]

    const int nodeBlocks = (NN + 255) / 256;   // 782
    const int edgeBlocks = (EE + 255) / 256;   // 25000
    const int gemmBlocks = (NN / 16) / 4;      // 3125 (4 waves/block, exact)

    init_kernel<<<nodeBlocks, 256, 0, stream>>>(dis, gsum, gcnt);
    deg_kernel<<<edgeBlocks, 256, 0, stream>>>(ecol, dis);
    dis_kernel<<<nodeBlocks, 256, 0, stream>>>(dis);

    gemm1_kernel<<<gemmBlocks, 128, 0, stream>>>(x, W1, b1, dis, h1, agg1);
    edge1_kernel<<<edgeBlocks, 256, 0, stream>>>(erow, ecol, dis, h1, agg1);

    gemm2_kernel<<<gemmBlocks, 128, 0, stream>>>(agg1, W2, b2, dis, h2, agg2);
    edge2_kernel<<<edgeBlocks, 256, 0, stream>>>(erow, ecol, dis, h2, agg2);

    pool_kernel<<<nodeBlocks, 256, 0, stream>>>(agg2, batch, gsum, gcnt);
    lsm_kernel<<<1, 64, 0, stream>>>(gsum, gcnt, out);
}